// Net_36240934043763
// MI455X (gfx1250) — compile-verified
//
#include <hip/hip_runtime.h>
#include <math.h>

// ---- problem constants (from reference) ----
#define BATCH   8
#define SEQ     512
#define IN_DIM  32
#define D_MODEL 512
#define N_LAYERS 4
#define D_INNER 1024
#define D_STATE 16
#define D_CONV  4
#define DT_RANK 32
#define M_ROWS  (BATCH*SEQ)      // 4096 token rows

typedef float v2f __attribute__((ext_vector_type(2)));
typedef float v8f __attribute__((ext_vector_type(8)));

// =====================================================================
// Register-blocked fp32 WMMA GEMM:
//   Y[M,N] (op)= X[M,K](row stride ldx) @ W[N,K]^T
// Each wave computes a 32x64 output tile (2x4 grid of 16x16 WMMA tiles):
// per k-step: 2 A-frag + 4 B-frag loads -> 8 v_wmma ops, ~100 VGPRs live
// (no accumulator spill). 4 waves/block tile along N => block tile 32x256.
// Requires: M % 32 == 0, N % 256 == 0, K % 4 == 0.
//   mode 0: Y = acc + bias[n]   (bias may be null)
//   mode 1: Y = softplus(acc + bias[n])
//   mode 2: Y += acc            (residual accumulate)
// =====================================================================
#define GEMM_WAVES 4
#define TM 2                      // 16-row tiles per wave (M)
#define TN 4                      // 16-col tiles per wave (N)
__global__ void gemm_blk_wmma_f32(const float* __restrict__ X, int ldx,
                                  const float* __restrict__ W,
                                  const float* __restrict__ bias,
                                  float* __restrict__ Y, int ldy,
                                  int K, int mode)
{
    const int lane = threadIdx.x & 31;
    const int wave = threadIdx.x >> 5;
    const int m0   = blockIdx.x * (16 * TM);
    const int n0   = (blockIdx.y * GEMM_WAVES + wave) * (16 * TN);
    const int half = lane >> 4;     // 0: holds K=0,1   1: holds K=2,3
    const int l16  = lane & 15;

    const float* aPtr[TM];
    const float* bPtr[TN];
    #pragma unroll
    for (int mi = 0; mi < TM; ++mi)
        aPtr[mi] = X + (size_t)(m0 + mi * 16 + l16) * ldx + 2 * half;
    #pragma unroll
    for (int ni = 0; ni < TN; ++ni)
        bPtr[ni] = W + (size_t)(n0 + ni * 16 + l16) * K + 2 * half;

    v8f acc[TM][TN] = {};
    for (int k = 0; k < K; k += 4) {
        v2f a[TM], b[TN];
        #pragma unroll
        for (int mi = 0; mi < TM; ++mi) { a[mi].x = aPtr[mi][k]; a[mi].y = aPtr[mi][k + 1]; }
        #pragma unroll
        for (int ni = 0; ni < TN; ++ni) { b[ni].x = bPtr[ni][k]; b[ni].y = bPtr[ni][k + 1]; }
        #pragma unroll
        for (int mi = 0; mi < TM; ++mi)
            #pragma unroll
            for (int ni = 0; ni < TN; ++ni)
                acc[mi][ni] = __builtin_amdgcn_wmma_f32_16x16x4_f32(
                    false, a[mi], false, b[ni], (short)0, acc[mi][ni], false, false);
    }

    #pragma unroll
    for (int ni = 0; ni < TN; ++ni) {
        const int n = n0 + ni * 16 + l16;
        const float bv = (mode != 2 && bias != nullptr) ? bias[n] : 0.f;
        #pragma unroll
        for (int mi = 0; mi < TM; ++mi) {
            #pragma unroll
            for (int r = 0; r < 8; ++r) {
                const int m = m0 + mi * 16 + r + 8 * half;
                float v = acc[mi][ni][r] + bv;
                float* yp = Y + (size_t)m * ldy + n;
                if (mode == 1) v = (v > 20.f) ? v : log1pf(expf(v));  // softplus
                if (mode == 2) *yp += v; else *yp = v;
            }
        }
    }
}

// ---- simple 16x16-per-wave WMMA GEMM (for small N, e.g. x_proj N=64) ----
__global__ void gemm16_wmma_f32(const float* __restrict__ X, int ldx,
                                const float* __restrict__ W,
                                const float* __restrict__ bias,
                                float* __restrict__ Y, int ldy,
                                int K, int mode)
{
    const int lane = threadIdx.x & 31;
    const int wave = threadIdx.x >> 5;
    const int m0   = blockIdx.x * 16;
    const int n0   = (blockIdx.y * GEMM_WAVES + wave) * 16;
    const int half = lane >> 4;
    const int l16  = lane & 15;

    const float* rowA = X + (size_t)(m0 + l16) * ldx + 2 * half;
    const float* rowB = W + (size_t)(n0 + l16) * K + 2 * half;

    v8f c = {};
    for (int k = 0; k < K; k += 4) {
        v2f a, b;
        a.x = rowA[k];  a.y = rowA[k + 1];
        b.x = rowB[k];  b.y = rowB[k + 1];
        c = __builtin_amdgcn_wmma_f32_16x16x4_f32(false, a, false, b,
                                                  (short)0, c, false, false);
    }

    const int   n  = n0 + l16;
    const float bv = (mode != 2 && bias != nullptr) ? bias[n] : 0.f;
    #pragma unroll
    for (int r = 0; r < 8; ++r) {
        const int m = m0 + r + 8 * half;
        float v = c[r] + bv;
        float* yp = Y + (size_t)m * ldy + n;
        if (mode == 1) v = (v > 20.f) ? v : log1pf(expf(v));
        if (mode == 2) *yp += v; else *yp = v;
    }
}

// ---------------- RMSNorm: one block per token row ----------------
__global__ void rmsnorm_k(const float* __restrict__ H,
                          const float* __restrict__ w,
                          float* __restrict__ O)
{
    __shared__ float red[256];
    const int row = blockIdx.x;
    const float* hp = H + (size_t)row * D_MODEL;
    float s = 0.f;
    for (int i = threadIdx.x; i < D_MODEL; i += 256) { float v = hp[i]; s += v * v; }
    red[threadIdx.x] = s; __syncthreads();
    for (int st = 128; st > 0; st >>= 1) {
        if (threadIdx.x < st) red[threadIdx.x] += red[threadIdx.x + st];
        __syncthreads();
    }
    const float scale = rsqrtf(red[0] / (float)D_MODEL + 1e-5f);
    float* op = O + (size_t)row * D_MODEL;
    for (int i = threadIdx.x; i < D_MODEL; i += 256) op[i] = hp[i] * scale * w[i];
}

// ------- causal depthwise conv (width 4) + SiLU over xr half of xz -------
__global__ void conv_silu_k(const float* __restrict__ XZ,      // [M_ROWS][2*D_INNER]
                            const float* __restrict__ cw,      // [D_INNER][4]
                            const float* __restrict__ cb,      // [D_INNER]
                            float* __restrict__ XC)            // [M_ROWS][D_INNER]
{
    int idx = blockIdx.x * blockDim.x + threadIdx.x;
    if (idx >= M_ROWS * D_INNER) return;
    const int d = idx & (D_INNER - 1);
    const int m = idx >> 10;                 // token row (b*SEQ + l)
    const int l = m & (SEQ - 1);             // position within sequence
    const float* wp = cw + d * D_CONV;
    float acc = cb[d];
    #pragma unroll
    for (int j = 0; j < D_CONV; ++j) {
        int lt = l - (D_CONV - 1) + j;
        if (lt >= 0)
            acc += wp[j] * XZ[(size_t)(m - (D_CONV - 1) + j) * (2 * D_INNER) + d];
    }
    XC[idx] = acc / (1.f + expf(-acc));      // SiLU
}

// ---------------- selective scan: lane per (b,d), 16 states in regs ----------------
__global__ void scan_k(const float* __restrict__ DELTA,  // [M_ROWS][D_INNER]
                       const float* __restrict__ XC,     // [M_ROWS][D_INNER]
                       const float* __restrict__ DBC,    // [M_ROWS][64]  (B at +32, C at +48)
                       const float* __restrict__ A_log,  // [D_INNER][16]
                       const float* __restrict__ Dp,     // [D_INNER]
                       float* __restrict__ Yo)           // [M_ROWS][D_INNER]
{
    __shared__ float Bs[32][D_STATE];
    __shared__ float Cs[32][D_STATE];
    const int b = blockIdx.y;
    const int d = blockIdx.x * blockDim.x + threadIdx.x;

    float A[D_STATE], h[D_STATE];
    #pragma unroll
    for (int n = 0; n < D_STATE; ++n) { A[n] = -expf(A_log[d * D_STATE + n]); h[n] = 0.f; }
    const float Dd = Dp[d];

    for (int t0 = 0; t0 < SEQ; t0 += 32) {
        // prefetch next chunk's per-thread stream while this chunk computes
        if (t0 + 32 < SEQ) {
            const size_t noff = (size_t)(b * SEQ + t0 + 32) * D_INNER + d;
            __builtin_prefetch(&DELTA[noff], 0, 0);   // -> global_prefetch_b8
            __builtin_prefetch(&XC[noff],    0, 0);
        }
        __syncthreads();
        for (int i = threadIdx.x; i < 32 * D_STATE; i += blockDim.x) {
            const int tt = i >> 4, n = i & 15;
            const float* drow = DBC + (size_t)(b * SEQ + t0 + tt) * (DT_RANK + 2 * D_STATE);
            Bs[tt][n] = drow[DT_RANK + n];
            Cs[tt][n] = drow[DT_RANK + D_STATE + n];
        }
        __syncthreads();
        for (int tt = 0; tt < 32; ++tt) {
            const size_t off = (size_t)(b * SEQ + t0 + tt) * D_INNER + d;
            const float dl = DELTA[off];
            const float xv = XC[off];
            float acc = 0.f;
            #pragma unroll
            for (int n = 0; n < D_STATE; ++n) {
                h[n] = expf(dl * A[n]) * h[n] + dl * Bs[tt][n] * xv;
                acc += h[n] * Cs[tt][n];
            }
            Yo[off] = acc + Dd * xv;
        }
    }
}

// ---------------- gating: y *= silu(z) ----------------
__global__ void gate_k(float* __restrict__ Y, const float* __restrict__ XZ)
{
    int idx = blockIdx.x * blockDim.x + threadIdx.x;
    if (idx >= M_ROWS * D_INNER) return;
    const int d = idx & (D_INNER - 1);
    const int m = idx >> 10;
    const float z = XZ[(size_t)m * (2 * D_INNER) + D_INNER + d];
    Y[idx] *= z / (1.f + expf(-z));
}

// ---------------- decoder (tiny: 8 rows) ----------------
__global__ void dec1_k(const float* __restrict__ H, const float* __restrict__ W1,
                       const float* __restrict__ b1, float* __restrict__ O)
{
    const int b = blockIdx.x;         // 8
    const int j = threadIdx.x;        // 256
    const float* hp = H + (size_t)(b * SEQ + SEQ - 1) * D_MODEL;
    const float* wp = W1 + (size_t)j * D_MODEL;
    float acc = b1[j];
    for (int k = 0; k < D_MODEL; ++k) acc += hp[k] * wp[k];
    O[b * (D_MODEL / 2) + j] = fmaxf(acc, 0.f);
}

__global__ void dec2_k(const float* __restrict__ H1, const float* __restrict__ W2,
                       const float* __restrict__ b2, float* __restrict__ OUT)
{
    const int b = threadIdx.x;
    if (b >= BATCH) return;
    float acc = b2[0];
    for (int k = 0; k < D_MODEL / 2; ++k) acc += H1[b * (D_MODEL / 2) + k] * W2[k];
    OUT[b] = acc;
}

// =====================================================================
extern "C" void kernel_launch(void* const* d_in, const int* in_sizes, int n_in,
                              void* d_out, int out_size, void* d_ws, size_t ws_size,
                              hipStream_t stream) {
    (void)in_sizes; (void)n_in; (void)out_size; (void)ws_size;

    const float* x        = (const float*)d_in[0];   // (8,512,32)
    const float* enc_w    = (const float*)d_in[1];   // (512,32)
    const float* enc_b    = (const float*)d_in[2];   // (512)
    const float* in_w     = (const float*)d_in[3];   // (4,2048,512)
    const float* conv_w   = (const float*)d_in[4];   // (4,1024,1,4)
    const float* conv_b   = (const float*)d_in[5];   // (4,1024)
    const float* xp_w     = (const float*)d_in[6];   // (4,64,1024)
    const float* dtp_w    = (const float*)d_in[7];   // (4,1024,32)
    const float* dtp_b    = (const float*)d_in[8];   // (4,1024)
    const float* A_log    = (const float*)d_in[9];   // (4,1024,16)
    const float* Dp       = (const float*)d_in[10];  // (4,1024)
    const float* out_w    = (const float*)d_in[11];  // (4,512,1024)
    const float* norm_w   = (const float*)d_in[12];  // (4,512)
    const float* dec_w1   = (const float*)d_in[13];  // (256,512)
    const float* dec_b1   = (const float*)d_in[14];  // (256)
    const float* dec_w2   = (const float*)d_in[15];  // (1,256)
    const float* dec_b2   = (const float*)d_in[16];  // (1)

    // workspace layout (floats)
    float* ws  = (float*)d_ws;
    float* H   = ws;                                   // 4096*512
    float* XN  = H   + (size_t)M_ROWS * D_MODEL;       // 4096*512
    float* XZ  = XN  + (size_t)M_ROWS * D_MODEL;       // 4096*2048
    float* XC  = XZ  + (size_t)M_ROWS * 2 * D_INNER;   // 4096*1024
    float* DBC = XC  + (size_t)M_ROWS * D_INNER;       // 4096*64
    float* DEL = DBC + (size_t)M_ROWS * 64;            // 4096*1024
    float* Yb  = DEL + (size_t)M_ROWS * D_INNER;       // 4096*1024
    float* H1  = Yb  + (size_t)M_ROWS * D_INNER;       // 8*256

    const dim3 gemmBlk(32 * GEMM_WAVES);
    const int  ew = M_ROWS * D_INNER;                  // elementwise total
    const dim3 ewGrid((ew + 255) / 256);
    const int  MB = 16 * TM;                           // 32-row block tile
    const int  NB = 16 * TN * GEMM_WAVES;              // 256-col block tile

    // encoder: H = x @ enc_w^T + enc_b    (M=4096, N=512, K=32)
    gemm_blk_wmma_f32<<<dim3(M_ROWS / MB, D_MODEL / NB), gemmBlk, 0, stream>>>(
        x, IN_DIM, enc_w, enc_b, H, D_MODEL, IN_DIM, 0);

    for (int i = 0; i < N_LAYERS; ++i) {
        // xn = rmsnorm(H) * norm_w[i]
        rmsnorm_k<<<M_ROWS, 256, 0, stream>>>(H, norm_w + (size_t)i * D_MODEL, XN);

        // xz = xn @ in_w[i]^T             (N=2048, K=512)
        gemm_blk_wmma_f32<<<dim3(M_ROWS / MB, (2 * D_INNER) / NB), gemmBlk, 0, stream>>>(
            XN, D_MODEL, in_w + (size_t)i * 2 * D_INNER * D_MODEL, nullptr,
            XZ, 2 * D_INNER, D_MODEL, 0);

        // xc = silu(causal_conv(xr) + cb)
        conv_silu_k<<<ewGrid, 256, 0, stream>>>(
            XZ, conv_w + (size_t)i * D_INNER * D_CONV, conv_b + (size_t)i * D_INNER, XC);

        // dbc = xc @ xp_w[i]^T            (N=64, K=1024) -- small N, simple kernel
        gemm16_wmma_f32<<<dim3(M_ROWS / 16, 64 / (16 * GEMM_WAVES)), gemmBlk, 0, stream>>>(
            XC, D_INNER, xp_w + (size_t)i * 64 * D_INNER, nullptr, DBC, 64, D_INNER, 0);

        // delta = softplus(dbc[:, :32] @ dtp_w[i]^T + dtp_b[i])   (N=1024, K=32)
        gemm_blk_wmma_f32<<<dim3(M_ROWS / MB, D_INNER / NB), gemmBlk, 0, stream>>>(
            DBC, 64, dtp_w + (size_t)i * D_INNER * DT_RANK, dtp_b + (size_t)i * D_INNER,
            DEL, D_INNER, DT_RANK, 1);

        // selective scan
        scan_k<<<dim3(D_INNER / 256, BATCH), 256, 0, stream>>>(
            DEL, XC, DBC, A_log + (size_t)i * D_INNER * D_STATE, Dp + (size_t)i * D_INNER, Yb);

        // y *= silu(z)
        gate_k<<<ewGrid, 256, 0, stream>>>(Yb, XZ);

        // H += y @ out_w[i]^T             (N=512, K=1024, residual accumulate)
        gemm_blk_wmma_f32<<<dim3(M_ROWS / MB, D_MODEL / NB), gemmBlk, 0, stream>>>(
            Yb, D_INNER, out_w + (size_t)i * D_MODEL * D_INNER, nullptr, H, D_MODEL, D_INNER, 2);
    }

    // decoder on last timestep of each batch
    dec1_k<<<BATCH, D_MODEL / 2, 0, stream>>>(H, dec_w1, dec_b1, H1);
    dec2_k<<<1, 64, 0, stream>>>(H1, dec_w2, dec_b2, (float*)d_out);
}